// QuantumMNIST_46299747451443
// MI455X (gfx1250) — compile-verified
//
#include <hip/hip_runtime.h>
#include <hip/hip_bf16.h>
#include <math.h>

#define N_WIRES 12
#define QDIM    4096          // 2^12
#define BATCH   2048
#define IN_DIM  784
#define PI_F    3.14159265358979323846f

typedef __attribute__((ext_vector_type(2))) float v2f;
typedef __attribute__((ext_vector_type(8))) float v8f;
typedef __attribute__((ext_vector_type(4))) int   v4i;

// pointer-to-int4 in global (AS1) / LDS (AS3), per the builtin's prototype
typedef __attribute__((address_space(1))) v4i global_v4i;
typedef __attribute__((address_space(3))) v4i lds_v4i;

#if __has_builtin(__builtin_amdgcn_global_load_async_to_lds_b128)
#define HAVE_ASYNC_LDS 1
#else
#define HAVE_ASYNC_LDS 0
#endif

// ---------------------------------------------------------------------------
// Stage 1: angles = tanh(x @ W1^T + b1) * pi via V_WMMA_F32_16X16X4_F32.
// W1 (12x784 f32 = 37.6 KB) is staged once per block into LDS with the
// gfx1250 async-to-LDS DMA path (ASYNCcnt), then shared by all 8 waves.
// One wave per 16-row tile of x (128 tiles, 8 waves/block, 16 blocks).
// A (16x4 f32): lane = M row (lane&15), half-wave selects K pair.
// B (4x16 f32): lane = N col (lane&15), read from LDS; cols >= 12 masked by
// multiply (no EXEC divergence around the WMMA).
// Two independent accumulator chains hide the WMMA D->C RAW latency.
// ---------------------------------------------------------------------------
__global__ __launch_bounds__(256)
void qm_gemm1_tanh(const float* __restrict__ x,
                   const float* __restrict__ W1,
                   const float* __restrict__ b1,
                   float* __restrict__ angles) {
  __shared__ float w1s[N_WIRES * IN_DIM];   // 9408 floats = 36.75 KB

  const int tid = threadIdx.x;

  // --- cooperative W1 -> LDS stage (async DMA on gfx1250) ---
#if HAVE_ASYNC_LDS
  for (int idx = tid; idx < (N_WIRES * IN_DIM) / 4; idx += 256) {
    __builtin_amdgcn_global_load_async_to_lds_b128(
        (global_v4i*)(W1 + idx * 4),
        (lds_v4i*)(w1s + idx * 4),
        0, 0);
  }
#if __has_builtin(__builtin_amdgcn_s_wait_asynccnt)
  __builtin_amdgcn_s_wait_asynccnt(0);
#else
  asm volatile("s_wait_asynccnt 0x0" ::: "memory");
#endif
#else
  for (int idx = tid; idx < N_WIRES * IN_DIM; idx += 256) w1s[idx] = W1[idx];
#endif
  __syncthreads();

  const int wave = tid >> 5;
  const int lane = tid & 31;
  const int tile = blockIdx.x * 8 + wave;       // 0..127
  const int m0   = tile * 16;
  const int mn   = lane & 15;                   // A row index / B col index
  const int half = lane >> 4;
  const int ksub = half * 2;

  const float* xrow   = x + (size_t)(m0 + mn) * IN_DIM;
  const int    nclamp = (mn < N_WIRES) ? mn : 0;
  const float  bmask  = (mn < N_WIRES) ? 1.0f : 0.0f;
  const float* w1p    = w1s + nclamp * IN_DIM;  // LDS row

  v8f acc0 = {};
  v8f acc1 = {};
  for (int k = 0; k < IN_DIM; k += 8) {
    __builtin_prefetch(xrow + k + 128, 0, 0);   // global_prefetch_b8, streaming x
    v2f a0, b0, a1, b1v;
    a0.x  = xrow[k + ksub];
    a0.y  = xrow[k + ksub + 1];
    b0.x  = w1p[k + ksub] * bmask;
    b0.y  = w1p[k + ksub + 1] * bmask;
    a1.x  = xrow[k + 4 + ksub];
    a1.y  = xrow[k + 4 + ksub + 1];
    b1v.x = w1p[k + 4 + ksub] * bmask;
    b1v.y = w1p[k + 4 + ksub + 1] * bmask;
    // (neg_a, A, neg_b, B, c_mod, C, reuse_a, reuse_b)
    acc0 = __builtin_amdgcn_wmma_f32_16x16x4_f32(false, a0, false, b0,
                                                 (short)0, acc0, false, false);
    acc1 = __builtin_amdgcn_wmma_f32_16x16x4_f32(false, a1, false, b1v,
                                                 (short)0, acc1, false, false);
  }
  const v8f acc = acc0 + acc1;

  if (mn < N_WIRES) {
    const float bias = b1[mn];
#pragma unroll
    for (int r = 0; r < 8; ++r) {
      const int row = m0 + r + half * 8;   // C/D layout: VGPR r -> M = r+half*8
      angles[row * N_WIRES + mn] = tanhf(acc[r] + bias) * PI_F;
    }
  }
}

// ---------------------------------------------------------------------------
// Stage 2: full 12-qubit circuit, one workgroup per sample, state in LDS.
// Wire w acts on index bit (11-w) (from psi.reshape(B, 2^w, 2, -1)).
// RY+RZ fused per pair update; CNOT = swap of target-bit pair where control=1.
// Finishes with <Z_w> reduction and the 12->10 linear head into d_out.
// ---------------------------------------------------------------------------
__global__ __launch_bounds__(256)
void qm_state_sim(const float* __restrict__ angles,
                  const float* __restrict__ ry_theta,
                  const float* __restrict__ rz_theta,
                  const float* __restrict__ W2,
                  const float* __restrict__ b2,
                  float* __restrict__ out) {
  __shared__ float re[QDIM];     // 16 KB
  __shared__ float im[QDIM];     // 16 KB
  __shared__ float zsh[N_WIRES];

  const int s   = blockIdx.x;
  const int tid = threadIdx.x;

  for (int i = tid; i < QDIM; i += 256) {
    re[i] = (i == 0) ? 1.0f : 0.0f;
    im[i] = 0.0f;
  }
  __syncthreads();

  // --- data-encoding RY gates (per-sample angles) ---
  for (int w = 0; w < N_WIRES; ++w) {
    const float th = 0.5f * angles[s * N_WIRES + w];
    const float cg = __cosf(th);
    const float sg = __sinf(th);
    const int   b      = 11 - w;
    const int   stride = 1 << b;
    for (int p = tid; p < QDIM / 2; p += 256) {
      const int i0 = ((p >> b) << (b + 1)) | (p & (stride - 1));
      const int i1 = i0 + stride;
      const float r0 = re[i0], r1 = re[i1];
      const float q0 = im[i0], q1 = im[i1];
      re[i0] = cg * r0 - sg * r1;  im[i0] = cg * q0 - sg * q1;
      re[i1] = sg * r0 + cg * r1;  im[i1] = sg * q0 + cg * q1;
    }
    __syncthreads();
  }

  // --- variational layers (shared ry/rz params) ---
  const float thy = 0.5f * ry_theta[0];
  const float thz = 0.5f * rz_theta[0];
  const float cy = __cosf(thy), sy = __sinf(thy);
  const float cz = __cosf(thz), sz = __sinf(thz);

  for (int layer = 0; layer < 3; ++layer) {
    for (int w = 0; w < N_WIRES; ++w) {        // fused RY then RZ on wire w
      const int b      = 11 - w;
      const int stride = 1 << b;
      for (int p = tid; p < QDIM / 2; p += 256) {
        const int i0 = ((p >> b) << (b + 1)) | (p & (stride - 1));
        const int i1 = i0 + stride;
        const float r0 = re[i0], r1 = re[i1];
        const float q0 = im[i0], q1 = im[i1];
        const float nr0 = cy * r0 - sy * r1, nq0 = cy * q0 - sy * q1;
        const float nr1 = sy * r0 + cy * r1, nq1 = sy * q0 + cy * q1;
        // bit=0 amp *= (cz - i sz); bit=1 amp *= (cz + i sz)
        re[i0] = nr0 * cz + nq0 * sz;  im[i0] = nq0 * cz - nr0 * sz;
        re[i1] = nr1 * cz - nq1 * sz;  im[i1] = nq1 * cz + nr1 * sz;
      }
      __syncthreads();
    }
    for (int w = 0; w < N_WIRES - 1; ++w) {    // CNOT(control=w, target=w+1)
      const int tb = 10 - w;                   // target bit; control bit = tb+1
      for (int q = tid; q < QDIM / 4; q += 256) {
        const int i = ((q >> tb) << (tb + 2)) | (2 << tb) | (q & ((1 << tb) - 1));
        const int j = i | (1 << tb);
        const float tr = re[i]; re[i] = re[j]; re[j] = tr;
        const float ti = im[i]; im[i] = im[j]; im[j] = ti;
      }
      __syncthreads();
    }
  }

  // --- measurement <Z_w> ---
  float part[N_WIRES];
#pragma unroll
  for (int w = 0; w < N_WIRES; ++w) part[w] = 0.0f;
  for (int i = tid; i < QDIM; i += 256) {
    const float p = re[i] * re[i] + im[i] * im[i];
#pragma unroll
    for (int w = 0; w < N_WIRES; ++w)
      part[w] += ((i >> (11 - w)) & 1) ? -p : p;
  }
  __syncthreads();                       // state dead; reuse re[] for reduction
#pragma unroll
  for (int w = 0; w < N_WIRES; ++w) re[tid * N_WIRES + w] = part[w];  // 3072 < 4096
  __syncthreads();
  if (tid < N_WIRES) {
    float z = 0.0f;
    for (int t = 0; t < 256; ++t) z += re[t * N_WIRES + tid];
    zsh[tid] = z;
  }
  __syncthreads();

  // --- logits = z @ W2^T + b2 ---
  if (tid < 10) {
    float acc = b2[tid];
#pragma unroll
    for (int n = 0; n < N_WIRES; ++n) acc += zsh[n] * W2[tid * N_WIRES + n];
    out[s * 10 + tid] = acc;
  }
}

extern "C" void kernel_launch(void* const* d_in, const int* in_sizes, int n_in,
                              void* d_out, int out_size, void* d_ws, size_t ws_size,
                              hipStream_t stream) {
  const float* x  = (const float*)d_in[0];
  const float* W1 = (const float*)d_in[1];
  const float* b1 = (const float*)d_in[2];
  const float* ry = (const float*)d_in[3];   // scalar (in_sizes==1)
  const float* rz = (const float*)d_in[4];   // scalar
  const float* W2 = (const float*)d_in[5];
  const float* b2 = (const float*)d_in[6];
  float* out    = (float*)d_out;             // 2048 x 10 fp32
  float* angles = (float*)d_ws;              // 2048 x 12 fp32 scratch (96 KB)

  qm_gemm1_tanh<<<BATCH / 16 / 8, 256, 0, stream>>>(x, W1, b1, angles);
  qm_state_sim<<<BATCH, 256, 0, stream>>>(angles, ry, rz, W2, b2, out);
}